// StructuralSelfAttention_8426725835121
// MI455X (gfx1250) — compile-verified
//
#include <hip/hip_runtime.h>
#include <hip/hip_bf16.h>

typedef __attribute__((ext_vector_type(2))) float v2f;
typedef __attribute__((ext_vector_type(4))) float v4f;
typedef __attribute__((ext_vector_type(8))) float v8f;

#define BATCH 64
#define SEQ   256
#define EMB   64
#define NQKV  64
#define LGRP  128
#define DIN   8192   // L*EMB
#define KT    32     // K-chunk per LDS stage
#define APAD  36     // row stride: 16B-aligned b128 stores, stride%32==4 -> bank-clean
#define BPAD  36

// ---- CDNA5 fp32 WMMA: D = A(16x4) * B(4x16) + C(16x16), full precision ----
__device__ __forceinline__ v8f wmma4(v2f a, v2f b, v8f c) {
  return __builtin_amdgcn_wmma_f32_16x16x4_f32(false, a, false, b, (short)0, c,
                                               false, false);
}

struct GemmArgs {
  const float* W[6];     // [g*3 + which], which: 0=Q,1=K,2=V
  const float* bias[6];
  const float* x;        // [B, SEQ, EMB]
  float*       out;      // ws: 6 matrices of [B, DIN]
};

// ================= Kernel 1: fused Q/K/V projections ======================
// grid (DIN/128, 6), 256 threads (8 waves as 4(M) x 2(N) of 32x32 tiles).
// Each block: 128 output features x all 64 batches -> every weight byte is
// read from HBM exactly once. Double-buffered LDS hides global latency
// behind the WMMA block; row-major tiles make every fragment one b64 read.
__global__ __launch_bounds__(256) void qkv_proj_kernel(GemmArgs p) {
  __shared__ float lA[2][128][APAD];   // weight tile  [o][k]
  __shared__ float lB[2][64][BPAD];    // xs tile      [b][k] (row-major!)

  const int m = blockIdx.y;            // which matrix (0..5)
  const int g = m / 3;
  const float* __restrict__ W    = p.W[m];
  const float* __restrict__ bias = p.bias[m];
  const float* __restrict__ x    = p.x;
  float* __restrict__ out = p.out + (size_t)m * BATCH * DIN;

  const int tid  = threadIdx.x;
  const int wave = tid >> 5;
  const int lane = tid & 31;
  const int l16  = lane & 15;
  const int khi  = (lane >> 4) << 1;   // 0 or 2: fragment K split
  const int mi   = wave & 3;           // M strip: mi*32
  const int ni   = wave >> 2;          // N strip: ni*32
  const int obase = blockIdx.x * 128;

  v8f acc[2][2] = {};                  // 32(M) x 32(N) per wave

  v4f wa[4], xb[2];                    // in-flight staging registers
  auto fetchChunk = [&](int k0) {
#pragma unroll
    for (int i = 0; i < 4; ++i) {      // A: 128 rows x 8 float4
      int idx = tid + i * 256, r = idx >> 3, c4 = idx & 7;
      wa[i] = *(const v4f*)(W + (size_t)(obase + r) * DIN + k0 + c4 * 4);
    }
#pragma unroll
    for (int i = 0; i < 2; ++i) {      // B: 64 rows x 8 float4
      int idx = tid + i * 256, b = idx >> 3, c4 = idx & 7;
      xb[i] = *(const v4f*)(x + (size_t)b * (SEQ * EMB) + g * (LGRP * EMB) +
                            k0 + c4 * 4);
    }
  };
  auto stageChunk = [&](int buf) {
#pragma unroll
    for (int i = 0; i < 4; ++i) {
      int idx = tid + i * 256, r = idx >> 3, c4 = idx & 7;
      *(v4f*)&lA[buf][r][c4 * 4] = wa[i];     // aligned ds_store_b128
    }
#pragma unroll
    for (int i = 0; i < 2; ++i) {
      int idx = tid + i * 256, b = idx >> 3, c4 = idx & 7;
      *(v4f*)&lB[buf][b][c4 * 4] = xb[i];
    }
  };

  fetchChunk(0);
  stageChunk(0);
  int cur = 0;
  for (int k0 = 0; k0 < DIN; k0 += KT) {
    __syncthreads();
    const bool more = (k0 + KT) < DIN;
    if (more) fetchChunk(k0 + KT);     // global loads in flight during compute

    const float* a0p = &lA[cur][mi * 32 + l16][khi];
    const float* a1p = &lA[cur][mi * 32 + 16 + l16][khi];
    const float* b0p = &lB[cur][ni * 32 + l16][khi];
    const float* b1p = &lB[cur][ni * 32 + 16 + l16][khi];
#pragma unroll
    for (int kk = 0; kk < KT; kk += 4) {
      v2f a0 = *(const v2f*)(a0p + kk);      // single ds_load_b64 each
      v2f a1 = *(const v2f*)(a1p + kk);
      v2f b0 = *(const v2f*)(b0p + kk);
      v2f b1 = *(const v2f*)(b1p + kk);
      acc[0][0] = wmma4(a0, b0, acc[0][0]);
      acc[0][1] = wmma4(a0, b1, acc[0][1]);
      acc[1][0] = wmma4(a1, b0, acc[1][0]);
      acc[1][1] = wmma4(a1, b1, acc[1][1]);
    }
    if (more) stageChunk(cur ^ 1);     // waits loadcnt after compute, not before
    cur ^= 1;
  }

  const int halfsel = (lane >> 4) << 3;
#pragma unroll
  for (int am = 0; am < 2; ++am)
#pragma unroll
    for (int bn = 0; bn < 2; ++bn)
#pragma unroll
      for (int r = 0; r < 8; ++r) {
        int o = obase + mi * 32 + am * 16 + r + halfsel;
        int batch = ni * 32 + bn * 16 + l16;
        out[(size_t)batch * DIN + o] = acc[am][bn][r] + bias[o];
      }
}

// ================= Kernel 2: attention per (batch, group) =================
#define QK_PAD 68                       // stride%64==4 -> conflict-free frags
#define VT_PAD 132
#define SC_PAD 132
#define OFF_KV (128 * QK_PAD)           // 8704
#define OFF_S  (2 * 128 * QK_PAD)       // 17408 (kv region holds 64*132=8448 too)
#define SMEM_FLOATS (2 * 128 * QK_PAD + 128 * SC_PAD)   // 34304 -> 137,216 B

__global__ __launch_bounds__(256) void attn_kernel(const float* __restrict__ qkv,
                                                   float* __restrict__ out) {
  extern __shared__ float smem[];
  float* qs = smem;             // Q [128][68]
  float* kv = smem + OFF_KV;    // K [128][68], later V^T [64][132]
  float* sc = smem + OFF_S;     // scores [128][132]

  const int b = blockIdx.x;
  const int g = blockIdx.y;
  const int tid  = threadIdx.x;
  const int wave = tid >> 5;
  const int lane = tid & 31;
  const int l16  = lane & 15;
  const int khi  = (lane >> 4) << 1;
  const int halfsel = (lane >> 4) << 3;

  const float* Qg = qkv + ((size_t)(g * 3 + 0) * BATCH + b) * DIN;
  const float* Kg = qkv + ((size_t)(g * 3 + 1) * BATCH + b) * DIN;
  const float* Vg = qkv + ((size_t)(g * 3 + 2) * BATCH + b) * DIN;

  // load Q and K (row-major, aligned b128 LDS stores)
#pragma unroll
  for (int i = 0; i < 8; ++i) {
    int idx = tid + i * 256;            // 2048 float4
    int r = idx >> 4, c4 = idx & 15;
    v4f q4 = *(const v4f*)(Qg + r * 64 + c4 * 4);
    v4f k4 = *(const v4f*)(Kg + r * 64 + c4 * 4);
    *(v4f*)&qs[r * QK_PAD + c4 * 4] = q4;
    *(v4f*)&kv[r * QK_PAD + c4 * 4] = k4;
  }
  __syncthreads();

  // scores S[i][j] = sum_e Q[i][e] K[j][e]  (16 queries/wave x 8 key tiles)
  {
    v8f acc[8] = {};
    const float* aRow = qs + (wave * 16 + l16) * QK_PAD + khi;
#pragma unroll
    for (int e = 0; e < EMB; e += 4) {
      v2f a = *(const v2f*)(aRow + e);
#pragma unroll
      for (int t = 0; t < 8; ++t) {
        v2f bb = *(const v2f*)(kv + (t * 16 + l16) * QK_PAD + khi + e);
        acc[t] = wmma4(a, bb, acc[t]);
      }
    }
#pragma unroll
    for (int t = 0; t < 8; ++t)
#pragma unroll
      for (int r = 0; r < 8; ++r) {
        int i = wave * 16 + r + halfsel;
        sc[i * SC_PAD + t * 16 + l16] = acc[t][r];
      }
  }
  __syncthreads();

  // overlap: stage V TRANSPOSED over K's LDS while 128 threads run softmax
#pragma unroll
  for (int i = 0; i < 8; ++i) {
    int idx = tid + i * 256;
    int j = idx & 127, c4 = idx >> 7;   // j = key row, c4 = e-quad
    v4f v4 = *(const v4f*)(Vg + j * 64 + c4 * 4);
#pragma unroll
    for (int u = 0; u < 4; ++u) kv[(c4 * 4 + u) * VT_PAD + j] = v4[u];
  }
  // softmax over the QUERY axis (axis=1): normalize each column j over i
  if (tid < 128) {
    const int j = tid;
    float mx = -3.402823466e+38f;
    for (int i = 0; i < 128; ++i) mx = fmaxf(mx, sc[i * SC_PAD + j]);
    float sum = 0.f;
    for (int i = 0; i < 128; ++i) {
      float e = __expf(sc[i * SC_PAD + j] - mx);
      sc[i * SC_PAD + j] = e;
      sum += e;
    }
    float inv = 1.0f / sum;
    for (int i = 0; i < 128; ++i) sc[i * SC_PAD + j] *= inv;
  }
  __syncthreads();

  // Z = attn @ V, scaled by 1/sqrt(64); B frags contiguous thanks to V^T
  {
    v8f acc[4] = {};
    const float* aRow = sc + (wave * 16 + l16) * SC_PAD + khi;
#pragma unroll 4
    for (int j = 0; j < 128; j += 4) {
      v2f a = *(const v2f*)(aRow + j);
#pragma unroll
      for (int n = 0; n < 4; ++n) {
        int col = n * 16 + l16;                        // output feature e
        v2f bb = *(const v2f*)(kv + col * VT_PAD + khi + j);
        acc[n] = wmma4(a, bb, acc[n]);
      }
    }
#pragma unroll
    for (int n = 0; n < 4; ++n)
#pragma unroll
      for (int r = 0; r < 8; ++r) {
        int i = wave * 16 + r + halfsel;
        int e = n * 16 + l16;
        out[((size_t)b * SEQ + g * LGRP + i) * NQKV + e] = acc[n][r] * 0.125f;
      }
  }
}

// ============================= launch =====================================
extern "C" void kernel_launch(void* const* d_in, const int* in_sizes, int n_in,
                              void* d_out, int out_size, void* d_ws, size_t ws_size,
                              hipStream_t stream) {
  (void)in_sizes; (void)n_in; (void)out_size; (void)ws_size;

  GemmArgs ga;
  ga.x = (const float*)d_in[0];
  for (int g = 0; g < 2; ++g)
    for (int w = 0; w < 3; ++w) {
      ga.W[g * 3 + w]    = (const float*)d_in[1 + g * 6 + w * 2];
      ga.bias[g * 3 + w] = (const float*)d_in[2 + g * 6 + w * 2];
    }
  float* qkv = (float*)d_ws;          // 6 * 64 * 8192 floats = 12 MB
  ga.out = qkv;

  qkv_proj_kernel<<<dim3(DIN / 128, 6), 256, 0, stream>>>(ga);
  attn_kernel<<<dim3(BATCH, 2), 256, SMEM_FLOATS * sizeof(float), stream>>>(
      qkv, (float*)d_out);
}